// CorrelationLoss_55817394979131
// MI455X (gfx1250) — compile-verified
//
#include <hip/hip_runtime.h>
#include <hip/hip_bf16.h>
#include <math.h>

// ---------------------------------------------------------------------------
// CorrelationLoss on MI455X (gfx1250, wave32):
//   miu = mean(E, axis=0);  G = E^T E  (M=N=256, K=8192, fp32 WMMA 16x16x4)
//   cov = G/B - miu miu^T;  loss = || cov - I ||_F
// All cross-block combines go through workspace (deterministic; no float
// atomics). Gram GEMM stages K-chunks through LDS with CDNA5 async
// global->LDS loads (ASYNCcnt), double-buffered.
// ---------------------------------------------------------------------------

typedef __attribute__((ext_vector_type(2))) float v2f;
typedef __attribute__((ext_vector_type(8))) float v8f;
typedef __attribute__((ext_vector_type(4))) int   v4i;

// async global->LDS builtin takes (addrspace(1) v4i*, addrspace(3) v4i*, imm, imm)
typedef __attribute__((address_space(1))) v4i* gv4i_p;
typedef __attribute__((address_space(3))) v4i* lv4i_p;

#define CL_B 8192
#define CL_W 256
#define CL_KC 32              // K-chunk rows staged in LDS
#define CL_PAD 80             // padded LDS row stride (floats): bank-safe

#if defined(__gfx1250__) && __has_builtin(__builtin_amdgcn_global_load_async_to_lds_b128)
#define USE_ASYNC_LDS 1
#else
#define USE_ASYNC_LDS 0
#endif

__device__ __forceinline__ void cl_wait_async(int pending) {
#if defined(__gfx1250__)
#if __has_builtin(__builtin_amdgcn_s_wait_asynccnt)
  if (pending == 0) __builtin_amdgcn_s_wait_asynccnt(0);
  else              __builtin_amdgcn_s_wait_asynccnt(2);
#else
  if (pending == 0) asm volatile("s_wait_asynccnt 0" ::: "memory");
  else              asm volatile("s_wait_asynccnt 2" ::: "memory");
#endif
#else
  (void)pending;
#endif
}

// ---------------- Kernel 1: partial column sums (coalesced) ----------------
// 64 blocks x 256 threads; block p sums rows [p*128, p*128+128) per column.
__global__ __launch_bounds__(256) void cl_col_sum_partial(const float* __restrict__ A,
                                                          float* __restrict__ partial) {
  const int j = threadIdx.x;                 // column 0..255
  const int p = blockIdx.x;                  // 0..63
  const float* base = A + (size_t)p * 128 * CL_W + j;
  float s = 0.0f;
#pragma unroll 8
  for (int r = 0; r < 128; ++r) s += base[(size_t)r * CL_W];
  partial[p * CL_W + j] = s;
}

// ---------------- Kernel 2: finalize means ---------------------------------
__global__ __launch_bounds__(256) void cl_col_mean_finalize(const float* __restrict__ partial,
                                                            float* __restrict__ miu) {
  const int j = threadIdx.x;
  float s = 0.0f;
#pragma unroll 8
  for (int p = 0; p < 64; ++p) s += partial[p * CL_W + j];
  miu[j] = s * (1.0f / (float)CL_B);
}

// ---------------- Kernel 3: Gram GEMM via fp32 WMMA ------------------------
// Grid: 16 blocks (4x4 blocks of 64x64 output). Block: 512 threads = 16 waves,
// one wave per 16x16 tile. K consumed in 32-row chunks double-buffered in LDS.
__device__ __forceinline__ void cl_stage_chunk(const float* __restrict__ A, int kbase,
                                               int cm0, int cn0,
                                               float (*As)[CL_PAD], float (*Bs)[CL_PAD],
                                               int t) {
  const int lk = t >> 4;           // 0..31 row within chunk
  const int lm = (t & 15) << 2;    // 0..60, float4 column
  const float* ga = A + (size_t)(kbase + lk) * CL_W + cm0 + lm;
  const float* gb = A + (size_t)(kbase + lk) * CL_W + cn0 + lm;
#if USE_ASYNC_LDS
  __builtin_amdgcn_global_load_async_to_lds_b128((gv4i_p)ga, (lv4i_p)&As[lk][lm], 0, 0);
  __builtin_amdgcn_global_load_async_to_lds_b128((gv4i_p)gb, (lv4i_p)&Bs[lk][lm], 0, 0);
#else
  const float4 va = *(const float4*)ga;
  const float4 vb = *(const float4*)gb;
  *(float4*)&As[lk][lm] = va;
  *(float4*)&Bs[lk][lm] = vb;
#endif
}

__global__ __launch_bounds__(512) void cl_gram_wmma(const float* __restrict__ A,
                                                    const float* __restrict__ miu,
                                                    float* __restrict__ tile_partial) {
  __shared__ __align__(16) float As[2][CL_KC][CL_PAD];
  __shared__ __align__(16) float Bs[2][CL_KC][CL_PAD];

  const int bi = blockIdx.x >> 2;            // output block row (64 rows)
  const int bj = blockIdx.x & 3;             // output block col (64 cols)
  const int cm0 = bi * 64;
  const int cn0 = bj * 64;

  const int t    = threadIdx.x;
  const int wave = t >> 5;                   // 0..15
  const int lane = t & 31;
  const int wm   = wave >> 2;                // tile row in block (0..3)
  const int wn   = wave & 3;                 // tile col in block (0..3)
  const int half = lane >> 4;                // upper/lower half-wave
  const int ln   = lane & 15;

  const int am = wm * 16 + ln;               // A-frag M index (block-local col of E)
  const int bn = wn * 16 + ln;               // B-frag N index

  v8f c = {};                                // 16x16 fp32 accumulator (8 VGPRs)

  cl_stage_chunk(A, 0, cm0, cn0, As[0], Bs[0], t);

  const int nchunks = CL_B / CL_KC;          // 256
  for (int chunk = 0; chunk < nchunks; ++chunk) {
    const int buf = chunk & 1;
    const bool more = (chunk + 1) < nchunks;
    if (more) cl_stage_chunk(A, (chunk + 1) * CL_KC, cm0, cn0, As[buf ^ 1], Bs[buf ^ 1], t);
#if USE_ASYNC_LDS
    cl_wait_async(more ? 2 : 0);             // own chunk complete; next may fly
#endif
    __syncthreads();                         // all waves' stages visible

#pragma unroll
    for (int k0 = 0; k0 < CL_KC; k0 += 4) {
      // A-frag 16x4 (ISA layout): lanes 0-15 hold K=k0,k0+1; lanes 16-31 K=k0+2,k0+3
      v2f a, b;
      a.x = As[buf][k0 + 2 * half + 0][am];
      a.y = As[buf][k0 + 2 * half + 1][am];
      // B-frag 4x16: rows striped across lanes; upper half-wave offset by 2 rows
      b.x = Bs[buf][k0 + 2 * half + 0][bn];
      b.y = Bs[buf][k0 + 2 * half + 1][bn];
      c = __builtin_amdgcn_wmma_f32_16x16x4_f32(
          /*neg_a=*/false, a, /*neg_b=*/false, b,
          /*c_mod=*/(short)0, c, /*reuse_a=*/false, /*reuse_b=*/false);
    }
    __syncthreads();                         // done reading buf before it is restaged
  }

  // Epilogue: diff = G/B - miu_i*miu_j - delta_ij; accumulate diff^2.
  // C/D layout: VGPR r -> row M = r + 8*half, col N = ln.
  const float invB = 1.0f / (float)CL_B;
  const int jcol = cn0 + bn;
  const float mj = miu[jcol];
  float acc = 0.0f;
#pragma unroll
  for (int r = 0; r < 8; ++r) {
    const int irow = cm0 + wm * 16 + r + 8 * half;
    const float d = c[r] * invB - miu[irow] * mj - ((irow == jcol) ? 1.0f : 0.0f);
    acc += d * d;
  }
  // wave32 reduction (deterministic fixed order)
#pragma unroll
  for (int off = 16; off > 0; off >>= 1) acc += __shfl_xor(acc, off, 32);
  if (lane == 0) tile_partial[blockIdx.x * 16 + wave] = acc;
}

// ---------------- Kernel 4: final deterministic reduce + sqrt --------------
__global__ __launch_bounds__(256) void cl_reduce_final(const float* __restrict__ tile_partial,
                                                       float* __restrict__ out) {
  __shared__ float s[256];
  const int t = threadIdx.x;
  s[t] = tile_partial[t];
  __syncthreads();
  for (int w = 128; w > 0; w >>= 1) {
    if (t < w) s[t] += s[t + w];
    __syncthreads();
  }
  if (t == 0) out[0] = sqrtf(s[0]);
}

// ---------------------------------------------------------------------------
extern "C" void kernel_launch(void* const* d_in, const int* in_sizes, int n_in,
                              void* d_out, int out_size, void* d_ws, size_t ws_size,
                              hipStream_t stream) {
  (void)in_sizes; (void)n_in; (void)out_size; (void)ws_size;
  const float* A = (const float*)d_in[0];   // embedding [8192,256] fp32
  // d_in[1] (label) unused by the reference math.

  float* ws          = (float*)d_ws;
  float* partial     = ws;                    // 64*256 floats
  float* miu         = ws + 64 * CL_W;        // 256 floats
  float* tile_part   = ws + 64 * CL_W + CL_W; // 256 floats

  cl_col_sum_partial <<<64, 256, 0, stream>>>(A, partial);
  cl_col_mean_finalize<<<1, 256, 0, stream>>>(partial, miu);
  cl_gram_wmma       <<<16, 512, 0, stream>>>(A, miu, tile_part);
  cl_reduce_final    <<<1, 256, 0, stream>>>(tile_part, (float*)d_out);
}